// GNN_cell_view_17205638988667
// MI455X (gfx1250) — compile-verified
//
#include <hip/hip_runtime.h>
#include <math.h>

typedef __attribute__((ext_vector_type(2))) float v2f;
typedef __attribute__((ext_vector_type(8))) float v8f;

#define BGR 128            // graphs per batch
static const int NSc[4] = {3000, 1500, 750, 375};

// ---------------- utility atomics ----------------
__device__ __forceinline__ void atomicMaxF(float* addr, float val) {
  if (val >= 0.0f) atomicMax((int*)addr, __float_as_int(val));
  else             atomicMin((unsigned int*)addr, __float_as_uint(val));
}

// ---------------- h0 = x * w_ln1^T + b ----------------
__global__ void k_init_lin(const float* __restrict__ x, const float* __restrict__ w,
                           const float* __restrict__ b, float* __restrict__ h, int N) {
  int t = blockIdx.x * blockDim.x + threadIdx.x;
  int n = t >> 4, d = t & 15;
  if (n >= N) return;
  h[t] = x[n] * w[d] + b[d];
}

// ---------------- ht = h @ W^T via V_WMMA_F32_16X16X4_F32 ----------------
// one wave32 per 16-node tile; K=16 -> 4 chained WMMAs
__global__ void k_transform(const float* __restrict__ h, const float* __restrict__ W,
                            float* __restrict__ ht, int N) {
  int wid  = (blockIdx.x * blockDim.x + threadIdx.x) >> 5;
  int lane = threadIdx.x & 31;
  int base = wid * 16;
  if (base >= N) return;
  int row = lane & 15;
  int kh  = lane >> 4;                 // 0: K 0,1 ; 1: K 2,3 within chunk
  v8f acc = {};
#pragma unroll
  for (int c = 0; c < 4; ++c) {
    int k0 = c * 4 + kh * 2;
    v2f a, bf;
    a.x  = h[(base + row) * 16 + k0];
    a.y  = h[(base + row) * 16 + k0 + 1];
    // B[k][n] = (W^T)[k][n] = W[n][k]
    bf.x = W[row * 16 + k0];
    bf.y = W[row * 16 + k0 + 1];
    acc = __builtin_amdgcn_wmma_f32_16x16x4_f32(false, a, false, bf, (short)0, acc,
                                                false, false);
  }
#pragma unroll
  for (int r = 0; r < 8; ++r)
    ht[(base + r + kh * 8) * 16 + row] = acc[r];
}

// ---------------- per-node attention scalars ----------------
__global__ void k_dots(const float* __restrict__ ht, const float* __restrict__ asrc,
                       const float* __restrict__ adst, float* __restrict__ hs,
                       float* __restrict__ hd, int N) {
  int n = blockIdx.x * blockDim.x + threadIdx.x;
  if (n >= N) return;
  float s = 0.f, d = 0.f;
#pragma unroll
  for (int k = 0; k < 16; ++k) {
    float v = ht[n * 16 + k];
    s += v * asrc[k];
    d += v * adst[k];
  }
  hs[n] = s;
  hd[n] = d;
}

__global__ void k_clear(float* __restrict__ mx, float* __restrict__ denom,
                        float* __restrict__ num, int N) {
  int n = blockIdx.x * blockDim.x + threadIdx.x;
  if (n >= N) return;
  mx[n]    = -3.4e38f;
  denom[n] = 0.f;
#pragma unroll
  for (int k = 0; k < 16; ++k) num[n * 16 + k] = 0.f;
}

__global__ void k_zero(float* __restrict__ p, int n) {
  int t = blockIdx.x * blockDim.x + threadIdx.x;
  if (t < n) p[t] = 0.f;
}

// ---------------- edge pass 1: segment max of leaky_relu scores ----------------
// items [0,E): real edges;  [E,E+N): self loops
__global__ void k_edge_max(const int* __restrict__ ei, const float* __restrict__ hs,
                           const float* __restrict__ hd, float* __restrict__ mx,
                           int E, int N) {
  int t = blockIdx.x * blockDim.x + threadIdx.x;
  if (t >= E + N) return;
  int s, d;
  if (t < E) { s = ei[t]; d = ei[E + t]; }
  else       { s = d = t - E; }
  float e = hs[s] + hd[d];
  e = (e >= 0.f) ? e : 0.2f * e;
  atomicMaxF(&mx[d], e);
}

// ---------------- edge pass 2: exp / weighted scatter-add ----------------
// one lane per (item, feature): coalesced ht reads + coalesced num atomics
__global__ void k_edge_acc(const int* __restrict__ ei, const float* __restrict__ hs,
                           const float* __restrict__ hd, const float* __restrict__ ht,
                           const float* __restrict__ mx, float* __restrict__ denom,
                           float* __restrict__ num, int E, int N) {
  int t = blockIdx.x * blockDim.x + threadIdx.x;
  int item = t >> 4, k = t & 15;
  if (item >= E + N) return;
  int s, d;
  if (item < E) { s = ei[item]; d = ei[E + item]; }
  else          { s = d = item - E; }
  float e = hs[s] + hd[d];
  e = (e >= 0.f) ? e : 0.2f * e;
  float ee = __expf(e - mx[d]);
  if (k == 0) atomicAdd(&denom[d], ee);
  atomicAdd(&num[d * 16 + k], ht[s * 16 + k] * ee);
}

// ---------------- finalize GAT: relu(num/denom + b), max-pool into clusters ----
__global__ void k_fin_pool(const float* __restrict__ num, const float* __restrict__ denom,
                           const float* __restrict__ bias, const int* __restrict__ cl,
                           float* __restrict__ pooled, int N) {
  int t = blockIdx.x * blockDim.x + threadIdx.x;
  int n = t >> 4, k = t & 15;
  if (n >= N) return;
  float v = num[t] / (denom[n] + 1e-16f) + bias[k];
  v = (v > 0.f) ? v : 0.f;   // relu => non-negative: uint atomicMax is order-correct
  atomicMax((unsigned int*)&pooled[cl[n] * 16 + k], __float_as_uint(v));
}

// ---------------- LayerNorm over D=16 ----------------
__global__ void k_ln(float* __restrict__ x, const float* __restrict__ g,
                     const float* __restrict__ b, int N) {
  int n = blockIdx.x * blockDim.x + threadIdx.x;
  if (n >= N) return;
  float v[16], mu = 0.f;
#pragma unroll
  for (int k = 0; k < 16; ++k) { v[k] = x[n * 16 + k]; mu += v[k]; }
  mu *= (1.f / 16.f);
  float var = 0.f;
#pragma unroll
  for (int k = 0; k < 16; ++k) { float dlt = v[k] - mu; var += dlt * dlt; }
  var *= (1.f / 16.f);
  float r = rsqrtf(var + 1e-5f);
#pragma unroll
  for (int k = 0; k < 16; ++k) x[n * 16 + k] = (v[k] - mu) * r * g[k] + b[k];
}

// ---------------- GraphNorm: per-graph per-feature sums (one block/graph) -----
__global__ void k_gn_stats(const float* __restrict__ x, float* __restrict__ stats,
                           int npg) {
  int gph = blockIdx.x;
  int f = threadIdx.x & 15, j = threadIdx.x >> 4;
  const float* xg = x + (size_t)gph * npg * 16;
  float s1 = 0.f, s2 = 0.f;
  for (int r = j; r < npg; r += 16) {
    float v = xg[r * 16 + f];
    s1 += v; s2 += v * v;
  }
  __shared__ float S1[256], S2[256];
  S1[threadIdx.x] = s1; S2[threadIdx.x] = s2;
  __syncthreads();
  if (j == 0) {
    for (int jj = 1; jj < 16; ++jj) { s1 += S1[f + 16 * jj]; s2 += S2[f + 16 * jj]; }
    stats[gph * 32 + f]      = s1;
    stats[gph * 32 + 16 + f] = s2;
  }
}

__global__ void k_gn_apply(float* __restrict__ x, const float* __restrict__ stats,
                           const float* __restrict__ g, const float* __restrict__ b,
                           const float* __restrict__ a, int npg, int N) {
  int t = blockIdx.x * blockDim.x + threadIdx.x;
  int n = t >> 4, k = t & 15;
  if (n >= N) return;
  int gph = n / npg;
  float inv_n = 1.f / (float)npg;
  float mu = stats[gph * 32 + k] * inv_n;
  float ex2 = stats[gph * 32 + 16 + k] * inv_n;
  float ak = a[k];
  // var of (x - a*mu) = E[x^2] - (2a - a^2) mu^2
  float var = ex2 - (2.f * ak - ak * ak) * mu * mu;
  x[t] = (x[t] - ak * mu) * rsqrtf(var + 1e-5f) * g[k] + b[k];
}

// ---------------- MLP head, f32 WMMA, K chunked by 4 ----------------
// z = relu(rep[128,6000] @ w1^T[6000,1024] + b1)
__global__ void k_mlp1(const float* __restrict__ rep, const float* __restrict__ w1,
                       const float* __restrict__ b1, float* __restrict__ z) {
  int wid  = (blockIdx.x * blockDim.x + threadIdx.x) >> 5;
  int lane = threadIdx.x & 31;
  int mtile = wid >> 6;          // 8 tiles of M=128
  int ntile = wid & 63;          // 64 tiles of N=1024
  int row = lane & 15, kh = lane >> 4;
  const float* arow = rep + (size_t)(mtile * 16 + row) * 6000;
  const float* brow = w1  + (size_t)(ntile * 16 + row) * 6000;
  v8f acc = {};
#pragma unroll 4
  for (int c = 0; c < 1500; ++c) {
    int k0 = c * 4 + kh * 2;
    v2f a = *(const v2f*)(arow + k0);
    v2f bf = *(const v2f*)(brow + k0);
    acc = __builtin_amdgcn_wmma_f32_16x16x4_f32(false, a, false, bf, (short)0, acc,
                                                false, false);
  }
  int ncol = ntile * 16 + row;
  float bias = b1[ncol];
#pragma unroll
  for (int r = 0; r < 8; ++r) {
    int m = mtile * 16 + r + kh * 8;
    float v = acc[r] + bias;
    z[m * 1024 + ncol] = (v > 0.f) ? v : 0.f;
  }
}

// out = relu(z[128,1024] @ w2^T[1024,256] + b2)
__global__ void k_mlp2(const float* __restrict__ z, const float* __restrict__ w2,
                       const float* __restrict__ b2, float* __restrict__ out) {
  int wid  = (blockIdx.x * blockDim.x + threadIdx.x) >> 5;
  int lane = threadIdx.x & 31;
  int mtile = wid >> 4;          // 8 tiles of M=128
  int ntile = wid & 15;          // 16 tiles of N=256
  int row = lane & 15, kh = lane >> 4;
  const float* arow = z  + (size_t)(mtile * 16 + row) * 1024;
  const float* brow = w2 + (size_t)(ntile * 16 + row) * 1024;
  v8f acc = {};
#pragma unroll 8
  for (int c = 0; c < 256; ++c) {
    int k0 = c * 4 + kh * 2;
    v2f a = *(const v2f*)(arow + k0);
    v2f bf = *(const v2f*)(brow + k0);
    acc = __builtin_amdgcn_wmma_f32_16x16x4_f32(false, a, false, bf, (short)0, acc,
                                                false, false);
  }
  int ncol = ntile * 16 + row;
  float bias = b2[ncol];
#pragma unroll
  for (int r = 0; r < 8; ++r) {
    int m = mtile * 16 + r + kh * 8;
    float v = acc[r] + bias;
    out[m * 256 + ncol] = (v > 0.f) ? v : 0.f;
  }
}

// ---------------- host launch ----------------
extern "C" void kernel_launch(void* const* d_in, const int* in_sizes, int n_in,
                              void* d_out, int out_size, void* d_ws, size_t ws_size,
                              hipStream_t stream) {
  const float* x      = (const float*)d_in[0];
  const float* w_ln1  = (const float*)d_in[1];
  const float* b_ln1  = (const float*)d_in[2];
  const float* gat_w  = (const float*)d_in[3];
  const float* gat_as = (const float*)d_in[4];
  const float* gat_ad = (const float*)d_in[5];
  const float* gat_b  = (const float*)d_in[6];
  const float* ln_g   = (const float*)d_in[7];
  const float* ln_b   = (const float*)d_in[8];
  const float* gn_g   = (const float*)d_in[9];
  const float* gn_b   = (const float*)d_in[10];
  const float* gn_a   = (const float*)d_in[11];
  const float* w1     = (const float*)d_in[12];
  const float* b1     = (const float*)d_in[13];
  const float* w2     = (const float*)d_in[14];
  const float* b2     = (const float*)d_in[15];
  const int* eis[3] = {(const int*)d_in[16], (const int*)d_in[17], (const int*)d_in[18]};
  const int* cls[3] = {(const int*)d_in[19], (const int*)d_in[20], (const int*)d_in[21]};
  int Es[3] = {in_sizes[16] / 2, in_sizes[17] / 2, in_sizes[18] / 2};

  int Nlv[4];
  for (int i = 0; i < 4; ++i) Nlv[i] = BGR * NSc[i];

  char* ws = (char*)d_ws;
  size_t big = (size_t)Nlv[0] * 16 * sizeof(float);   // 24.576 MB
  float* bufA  = (float*)ws;                          // current h / pool target
  float* bufB  = (float*)(ws + big);                  // ht
  float* num   = (float*)(ws + 2 * big);              // attention numerator
  float* hs    = (float*)(ws + 3 * big);
  float* hd    = hs + Nlv[0];
  float* mx    = hd + Nlv[0];
  float* denom = mx + Nlv[0];
  float* stats = denom + Nlv[0];                      // 128*32
  float* zbuf  = stats + BGR * 32;                    // 128*1024

  auto cdiv = [](long long a, long long b) { return (int)((a + b - 1) / b); };
  const int T = 256;

  k_init_lin<<<cdiv((long long)Nlv[0] * 16, T), T, 0, stream>>>(x, w_ln1, b_ln1, bufA, Nlv[0]);

  for (int i = 0; i < 3; ++i) {
    int Ni = Nlv[i], Npo = Nlv[i + 1], E = Es[i];
    k_transform<<<cdiv((long long)Ni * 2, T), T, 0, stream>>>(bufA, gat_w + i * 256, bufB, Ni);
    k_dots<<<cdiv(Ni, T), T, 0, stream>>>(bufB, gat_as + i * 16, gat_ad + i * 16, hs, hd, Ni);
    k_clear<<<cdiv(Ni, T), T, 0, stream>>>(mx, denom, num, Ni);
    k_edge_max<<<cdiv((long long)E + Ni, T), T, 0, stream>>>(eis[i], hs, hd, mx, E, Ni);
    k_edge_acc<<<cdiv(((long long)E + Ni) * 16, T), T, 0, stream>>>(eis[i], hs, hd, bufB,
                                                                    mx, denom, num, E, Ni);
    k_zero<<<cdiv((long long)Npo * 16, T), T, 0, stream>>>(bufA, Npo * 16);
    k_fin_pool<<<cdiv((long long)Ni * 16, T), T, 0, stream>>>(num, denom, gat_b + i * 16,
                                                              cls[i], bufA, Ni);
    k_ln<<<cdiv(Npo, T), T, 0, stream>>>(bufA, ln_g + i * 16, ln_b + i * 16, Npo);
    k_gn_stats<<<BGR, T, 0, stream>>>(bufA, stats, NSc[i + 1]);
    k_gn_apply<<<cdiv((long long)Npo * 16, T), T, 0, stream>>>(bufA, stats, gn_g + i * 16,
                                                               gn_b + i * 16, gn_a + i * 16,
                                                               NSc[i + 1], Npo);
  }

  // bufA is now [128*375, 16] == rep [128, 6000] (contiguous reshape)
  k_mlp1<<<64, T, 0, stream>>>(bufA, w1, b1, zbuf);          // 512 waves
  k_mlp2<<<16, T, 0, stream>>>(zbuf, w2, b2, (float*)d_out); // 128 waves
}